// LinearFilter_8443905704514
// MI455X (gfx1250) — compile-verified
//
#include <hip/hip_runtime.h>

typedef float        v2f __attribute__((ext_vector_type(2)));
typedef float        v8f __attribute__((ext_vector_type(8)));
typedef unsigned int v4u __attribute__((ext_vector_type(4)));
typedef int          v4i __attribute__((ext_vector_type(4)));
typedef int          v8i __attribute__((ext_vector_type(8)));

namespace {
constexpr int kB = 8, kH = 512, kW = 512, kC = 32, kP = 16;
constexpr int kFiltN = kB * kH * kW;  // 2,097,152 filtered outputs
}

// Block: 256 threads = 8 waves. Covers batch b, h%16 == ph, rows
// h = ph + 16*(2*kChunk + {0,1}), all 512 columns -> 1024 pixels = 64 groups
// of 16 pixels. Each wave owns 16 pixels per iteration in WMMA A-layout:
//   lane = (pixel m = lane&15, half = lane>>4)
//   low half streams channels {4j,4j+1}, high half {4j+2,4j+3}  (K-chunks)
__global__ __launch_bounds__(256) void linfilt_kernel(
    const float* __restrict__ x, const float* __restrict__ wfil,
    const float* __restrict__ bias, float* __restrict__ out) {
  __shared__ float lds_f[kP * kC];  // 2 KB: filter slice w[0, ph, 0:16, 0:32]

  const int t        = threadIdx.x;
  const unsigned bid = blockIdx.x;
  const int kChunk   = bid & 15;         // which pair of rows in the h%16 class
  const int ph       = (bid >> 4) & 15;  // h % 16 for every pixel in this block
  const int b        = bid >> 8;         // batch index

  // ---- Stage the 2 KB filter slice into LDS via the Tensor Data Mover ----
#if __has_builtin(__builtin_amdgcn_tensor_load_to_lds) && __has_builtin(__builtin_amdgcn_s_wait_tensorcnt)
  if (t == 0) {
    unsigned long long ga = (unsigned long long)(const void*)wfil +
                            (unsigned long long)ph * (kP * kC * 4);
    unsigned lds_off = (unsigned)(unsigned long long)(&lds_f[0]);  // low 32 bits = LDS byte offset

    v4u g0;
    g0.x = 0x1u;                                // count=1 (valid user descriptor)
    g0.y = lds_off;                             // lds_addr
    g0.z = (unsigned)ga;                        // global_addr[31:0]
    g0.w = (unsigned)(ga >> 32) | (2u << 30);   // global_addr[56:32] | type=2 ("image")

    v8i g1;
    g1[0] = (2 << 16);    // data_size = 2 -> 4-byte elements; no multicast/pad/iterate
    g1[1] = (512 << 16);  // tensor_dim0[15:0] = 512 dwords
    g1[2] = (1 << 16);    // tensor_dim0[31:16]=0 | tensor_dim1[15:0] = 1
    g1[3] = (512 << 16);  // tensor_dim1[31:16]=0 | tile_dim0 = 512
    g1[4] = 1;            // tile_dim1 = 1, tile_dim2 = 0
    g1[5] = 512;          // tensor_dim0_stride[31:0]
    g1[6] = (512 << 16);  // tensor_dim0_stride[47:32]=0 | tensor_dim1_stride[15:0]
    g1[7] = 0;            // tensor_dim1_stride[47:16]

    v4i gz = {0, 0, 0, 0};
#if defined(__clang_major__) && __clang_major__ >= 23
    v8i gz8 = {0, 0, 0, 0, 0, 0, 0, 0};
    __builtin_amdgcn_tensor_load_to_lds(g0, g1, gz, gz, gz8, 0);
#else
    __builtin_amdgcn_tensor_load_to_lds(g0, g1, gz, gz, 0);
#endif
    __builtin_amdgcn_s_wait_tensorcnt(0);  // wave0: TDM landed in LDS
  }
  __syncthreads();
#else
  // Fallback: cooperative plain copy (512 dwords, 256 threads)
  lds_f[t]       = wfil[ph * (kP * kC) + t];
  lds_f[t + 256] = wfil[ph * (kP * kC) + t + 256];
  __syncthreads();
#endif

  const int lane = t & 31;
  const int wv   = t >> 5;      // wave id within block: 0..7
  const int pxl  = lane & 15;   // pixel-in-group == w%16 (filter column)
  const int half = lane >> 4;   // 0 -> channels {0,1} mod 4 ; 1 -> {2,3} mod 4

  // Per-lane filter fragment (16 floats) already in WMMA A-matrix layout.
  v2f f2[8];
  {
    const v2f* lf2 = (const v2f*)lds_f + (size_t)pxl * 16 + half;
#pragma unroll
    for (int j = 0; j < 8; ++j) f2[j] = lf2[2 * j];
  }
  const float bv = bias[ph * kP + pxl];  // bias for the pixel this lane stores

  float* filtered = out;           // [B*H*W]
  float* weighted = out + kFiltN;  // [B*H*W*C]

  const v2f ones = {1.f, 1.f};     // B matrix: 4x16 of 1.0 -> D columns = row sums
  const size_t lo2 = (size_t)pxl * 16 + half;  // per-lane offset, v2f units

  for (int i = 0; i < 8; ++i) {
    const int g     = wv + 8 * i;     // group 0..63
    const int r     = g >> 5;         // row within the block's pair
    const int wbase = (g & 31) * kP;  // group's first column
    const int k     = kChunk * 2 + r;
    const int h     = ph + kP * k;
    const bool doRelu = (k == 0) && (wbase == 0);  // top-left 16x16 image tile
    const size_t rowpix = ((size_t)b * kH + h) * kW + wbase;

    const v2f* xp = (const v2f*)(x + rowpix * kC) + lo2;
    v2f*       wp = (v2f*)(weighted + rowpix * kC) + lo2;

#if __has_builtin(__builtin_amdgcn_wmma_f32_16x16x4_f32)
    v8f d = {0.f, 0.f, 0.f, 0.f, 0.f, 0.f, 0.f, 0.f};
#pragma unroll
    for (int j = 0; j < 8; ++j) {
      v2f xv = __builtin_nontemporal_load(xp + 2 * j);  // streaming read
      v2f fv = f2[j];
      if (doRelu) {
        fv.x = fv.x > 0.f ? fv.x : 0.f;
        fv.y = fv.y > 0.f ? fv.y : 0.f;
      }
      v2f p = xv * fv;
      __builtin_nontemporal_store(p, wp + 2 * j);       // streaming write
      // D += A(products, 16x4) * B(ones, 4x16): every D column = K-chunk row sum
      d = __builtin_amdgcn_wmma_f32_16x16x4_f32(false, p, false, ones,
                                                (short)0, d, false, false);
    }
    // d[r]: lanes 0-15 = sum(pixel r), lanes 16-31 = sum(pixel r+8); columns equal.
    const int li = lane & 7;
    float v = d[0];
#pragma unroll
    for (int rr = 1; rr < 8; ++rr) v = (li == rr) ? d[rr] : v;
    float sw = __shfl_xor(v, 16, 32);   // bring high-half rows 8..15 to lanes 8..15
    if (lane & 8) v = sw;
    if (lane < 16)
      __builtin_nontemporal_store(v + bv, filtered + rowpix + lane);
#else
    // Fallback: VALU reduction in the same layout
    v2f acc = {0.f, 0.f};
#pragma unroll
    for (int j = 0; j < 8; ++j) {
      v2f xv = __builtin_nontemporal_load(xp + 2 * j);
      v2f fv = f2[j];
      if (doRelu) {
        fv.x = fv.x > 0.f ? fv.x : 0.f;
        fv.y = fv.y > 0.f ? fv.y : 0.f;
      }
      v2f p = xv * fv;
      __builtin_nontemporal_store(p, wp + 2 * j);
      acc += p;
    }
    float partial = acc.x + acc.y;                 // this half's 16-channel sum
    float cross = __shfl_xor(partial, 16, 32);     // the other half's sum
    if (lane < 16)
      __builtin_nontemporal_store(partial + cross + bv, filtered + rowpix + lane);
#endif
  }
}

extern "C" void kernel_launch(void* const* d_in, const int* in_sizes, int n_in,
                              void* d_out, int out_size, void* d_ws, size_t ws_size,
                              hipStream_t stream) {
  const float* x = (const float*)d_in[0];  // [8,512,512,32]
  const float* w = (const float*)d_in[1];  // [1,16,16,32]
  const float* b = (const float*)d_in[2];  // [1,16,16,1]
  float* out = (float*)d_out;              // filtered (2M) ++ weighted (64M)

  dim3 grid(kB * 16 * 16);  // 2048 blocks: b * (h%16) * rowChunk
  linfilt_kernel<<<grid, 256, 0, stream>>>(x, w, b, out);
}